// CustomPuzzleLoss_61718680043612
// MI455X (gfx1250) — compile-verified
//
#include <hip/hip_runtime.h>

#define PGRID 5
#define PELEMS 25

typedef __attribute__((ext_vector_type(2))) float v2f;
typedef __attribute__((ext_vector_type(8))) float v8f;

// Reduce one f32 value across the 32 lanes of a wave using
// V_WMMA_F32_16X16X4_F32 (full f32 precision).
//   A (16x4) = all ones, B (4x16) holds the 32 lane values in 32 of its 64
//   slots (b[1] = 0 fills the rest). D[m][n] = sum_k B[k][n], so every D row
//   is the vector of B column-sums. Documented C/D layout: row M=0 sits in
//   D VGPR0 lanes 0..15; summing those 16 lanes yields the wave total.
// EXEC must be all ones at the call site (call outside divergent control flow).
__device__ __forceinline__ float wave_sum_wmma(float v) {
    v2f a; a[0] = 1.0f; a[1] = 1.0f;   // A[m][k] = 1 everywhere
    v2f b; b[0] = v;    b[1] = 0.0f;   // one B slot per lane carries v
    v8f c = {};
    c = __builtin_amdgcn_wmma_f32_16x16x4_f32(
            /*neg_a=*/false, a, /*neg_b=*/false, b,
            /*c_mod=*/(short)0, c, /*reuse_a=*/false, /*reuse_b=*/false);
    float s = c[0];                    // D[0][lane], lanes 0..15
    s += __shfl_down(s, 8, 16);
    s += __shfl_down(s, 4, 16);
    s += __shfl_down(s, 2, 16);
    s += __shfl_down(s, 1, 16);
    return s;                          // valid in lane 0 (and lane 16)
}

// Duplicate count for one 5x5 grid held in registers: position j (j>=1) of a
// row/column counts once if ANY earlier position k<j holds an equal value
// (matches jnp.any over the strictly-lower triangle), for rows and columns.
__device__ __forceinline__ int dup_count_grid(const float* p) {
    int dup = 0;
#pragma unroll
    for (int r = 0; r < PGRID; ++r) {
#pragma unroll
        for (int j = 1; j < PGRID; ++j) {
            bool drow = false, dcol = false;
#pragma unroll
            for (int k = 0; k < j; ++k) {
                drow = drow || (p[r * PGRID + j] == p[r * PGRID + k]);
                dcol = dcol || (p[j * PGRID + r] == p[k * PGRID + r]);
            }
            dup += (int)drow + (int)dcol;
        }
    }
    return dup;
}

__global__ void puzzle_init_ws(float* ws_sum, unsigned int* ws_oob) {
    ws_sum[0] = 0.0f;   // sum |pred - target|
    ws_sum[1] = 0.0f;   // total duplicate count
    *ws_oob   = 0u;     // out-of-bounds flag
}

__global__ void __launch_bounds__(256, 1)
puzzle_loss_main(const float* __restrict__ preds,
                 const int*   __restrict__ targets,
                 float*       __restrict__ ws_sum,
                 unsigned int* __restrict__ ws_oob,
                 int num_grids) {
    const int tid    = blockIdx.x * blockDim.x + threadIdx.x;
    const int stride = gridDim.x * blockDim.x;

    float sum_abs = 0.0f;
    float dupf    = 0.0f;
    float vmin    =  1.0e30f;   // running min/max for the OOB test
    float vmax    = -1.0e30f;

    // Main path: 4 consecutive grids per thread = 400 bytes, 16B-aligned,
    // streamed with 25 b128 pred loads + 25 b128 target loads (max bytes in
    // flight per LOADcnt slot; fully coalesced).
    const int num_groups = num_grids >> 2;
    for (int grp = tid; grp < num_groups; grp += stride) {
        const float4* gp4 = reinterpret_cast<const float4*>(preds)   + (size_t)grp * 25;
        const int4*   gt4 = reinterpret_cast<const int4*>(targets)   + (size_t)grp * 25;

        float p[4 * PELEMS];
#pragma unroll
        for (int i = 0; i < 25; ++i) {
            const float4 q = gp4[i];
            p[4 * i + 0] = q.x; p[4 * i + 1] = q.y;
            p[4 * i + 2] = q.z; p[4 * i + 3] = q.w;
        }

#pragma unroll
        for (int i = 0; i < 25; ++i) {
            const int4 tq = gt4[i];
            sum_abs += fabsf(p[4 * i + 0] - (float)tq.x);
            sum_abs += fabsf(p[4 * i + 1] - (float)tq.y);
            sum_abs += fabsf(p[4 * i + 2] - (float)tq.z);
            sum_abs += fabsf(p[4 * i + 3] - (float)tq.w);
        }

#pragma unroll
        for (int i = 0; i < 4 * PELEMS; ++i) {
            vmin = fminf(vmin, p[i]);
            vmax = fmaxf(vmax, p[i]);
        }

        int dup = 0;
#pragma unroll
        for (int s = 0; s < 4; ++s)
            dup += dup_count_grid(p + PELEMS * s);
        dupf += (float)dup;
    }

    // Tail: leftover grids when num_grids % 4 != 0 (no-op for these sizes).
    for (int g = (num_groups << 2) + tid; g < num_grids; g += stride) {
        const float* gp = preds   + (size_t)g * PELEMS;
        const int*   gt = targets + (size_t)g * PELEMS;
        float p[PELEMS];
#pragma unroll
        for (int i = 0; i < PELEMS; ++i) p[i] = gp[i];
#pragma unroll
        for (int i = 0; i < PELEMS; ++i) {
            sum_abs += fabsf(p[i] - (float)gt[i]);
            vmin = fminf(vmin, p[i]);
            vmax = fmaxf(vmax, p[i]);
        }
        dupf += (float)dup_count_grid(p);
    }

    const int oob = (vmin < 0.5f) | (vmax > 5.5f);

    // All lanes reconverged: EXEC all ones -> WMMA legal.
    float wsum = wave_sum_wmma(sum_abs);
    float wdup = wave_sum_wmma(dupf);
    int any_oob = __any(oob);

    if ((threadIdx.x & 31) == 0) {
        atomicAdd(ws_sum + 0, wsum);
        atomicAdd(ws_sum + 1, wdup);
        if (any_oob) atomicOr(ws_oob, 1u);
    }
}

__global__ void puzzle_finalize(const float* __restrict__ ws_sum,
                                const unsigned int* __restrict__ ws_oob,
                                float* __restrict__ out,
                                int n, int num_grids) {
    float loss1 = ws_sum[0] / (float)n;
    float loss2 = ws_sum[1] / (float)num_grids * 0.1f;
    float pen   = (*ws_oob) ? 1000.0f : 0.0f;
    out[0] = loss1 + loss2 + pen;
}

extern "C" void kernel_launch(void* const* d_in, const int* in_sizes, int n_in,
                              void* d_out, int out_size, void* d_ws, size_t ws_size,
                              hipStream_t stream) {
    const float* preds   = (const float*)d_in[0];
    const int*   targets = (const int*)d_in[1];
    float*       out     = (float*)d_out;

    const int n         = in_sizes[0];
    const int num_grids = n / PELEMS;

    float*        ws_sum = (float*)d_ws;
    unsigned int* ws_oob = (unsigned int*)((char*)d_ws + 2 * sizeof(float));

    puzzle_init_ws<<<1, 1, 0, stream>>>(ws_sum, ws_oob);

    const int threads = 256;   // 8 wave32s per block
    const int blocks  = 1024;  // 262,144 threads: one 4-grid group per thread
    puzzle_loss_main<<<blocks, threads, 0, stream>>>(preds, targets, ws_sum,
                                                     ws_oob, num_grids);

    puzzle_finalize<<<1, 1, 0, stream>>>(ws_sum, ws_oob, out, n, num_grids);
}